// Decoder_42468636623472
// MI455X (gfx1250) — compile-verified
//
#include <hip/hip_runtime.h>
#include <hip/hip_bf16.h>
#include <math.h>

// ---------------------------------------------------------------------------
// Autoregressive LSTM decoder, H=1024 B=128 T=256 OUT=1 IN=2, full f32.
// Per step: gates = hT @ W_hh^T (+ inp@W_ih^T + bias) -> LSTM pointwise ->
// linear head -> feedback.  Sequential over T; each step is one kernel launch
// (graph-captured), dominant GEMM done with V_WMMA_F32_16X16X4_F32.
//
// WMMA mapping (per ISA 7.12.2, 32-bit fragments, wave32):
//   D tile 16x16: M = gate row (j within tile), N = batch.
//     lane l, vgpr v -> M = v + 8*(l>=16), N = l%16
//   A tile 16x4 (W_hh rows): lane l -> M=l%16 ; vgpr v -> K = v + 2*(l>=16)
//   B tile 4x16  (h^T)     : lane l -> N=l%16 ; vgpr v -> K = v + 2*(l>=16)
//
// Workspace (floats):
//   Apack : 4096*1024      W_hh pre-swizzled into A-fragment order (16 MB)
//   hTa/hTb: 1024*128 each transposed recurrent h, k-quad interleaved layout
//             (k,b) -> (k>>2)*512 + b*4 + (k&3)      (ping-pong per step)
//   cT    : 1024*128       cell state, (j,b) -> j*128 + b  (matches D layout)
//   combo : 4096*4         {b_ih+b_hh, W_ih[:,0], W_ih[:,1], 0} per gate row
// ---------------------------------------------------------------------------

#define HID    1024
#define BATCH  128
#define TSTEPS 256

typedef float v2f __attribute__((ext_vector_type(2)));
typedef float v4f __attribute__((ext_vector_type(4)));
typedef float v8f __attribute__((ext_vector_type(8)));

__device__ __forceinline__ v8f wmma4(v2f a, v2f b, v8f c) {
    // 8 args: (neg_a, A, neg_b, B, c_mod, C, reuse_a, reuse_b)
    return __builtin_amdgcn_wmma_f32_16x16x4_f32(false, a, false, b,
                                                 (short)0, c, false, false);
}

__device__ __forceinline__ float sigm(float x) {
    return 1.0f / (1.0f + __expf(-x));
}

// ---- one-time weight repack: W_hh -> A-fragment order, fuse biases/W_ih ----
__global__ __launch_bounds__(256) void repack_kernel(
    const float* __restrict__ W_hh, const float* __restrict__ W_ih,
    const float* __restrict__ b_ih, const float* __restrict__ b_hh,
    float* __restrict__ Apack, float* __restrict__ combo)
{
    const int idx = blockIdx.x * blockDim.x + threadIdx.x;  // 0 .. 1048575
    const int lane = idx & 31;
    const int q    = (idx >> 5) & 3;     // gate quadrant i,f,g,o
    const int kp   = (idx >> 7) & 127;   // k-pair (2 WMMA k-steps)
    const int jt   = idx >> 14;          // j tile 0..63
    const int lo   = lane & 15;
    const int hi   = lane >> 4;
    const int g    = q * HID + jt * 16 + lo;      // W_hh row (M = lane%16)
    v4f val;
#pragma unroll
    for (int c = 0; c < 4; ++c) {
        const int kstep = 2 * kp + (c >> 1);
        const int k     = 4 * kstep + (c & 1) + 2 * hi;   // A frag K slot
        val[c] = W_hh[g * HID + k];
    }
    *(v4f*)(Apack + (size_t)idx * 4) = val;

    if (idx < 4 * HID) {
        combo[idx * 4 + 0] = b_ih[idx] + b_hh[idx];
        combo[idx * 4 + 1] = W_ih[idx * 2 + 0];
        combo[idx * 4 + 2] = W_ih[idx * 2 + 1];
        combo[idx * 4 + 3] = 0.0f;
    }
}

// ---- one-time state init: transpose h0/c0, seed d_out with b_lin ----------
__global__ __launch_bounds__(256) void init_state_kernel(
    const float* __restrict__ h0, const float* __restrict__ c0,
    const float* __restrict__ b_lin, float* __restrict__ hT,
    float* __restrict__ cT, float* __restrict__ out)
{
    const int idx = blockIdx.x * blockDim.x + threadIdx.x;  // 0 .. 131071
    const int bb  = idx >> 10;
    const int k   = idx & 1023;
    hT[(k >> 2) * (BATCH * 4) + bb * 4 + (k & 3)] = h0[idx];  // (1,B,H) -> hT
    cT[k * BATCH + bb] = c0[idx];
    if (idx < BATCH * TSTEPS) out[idx] = b_lin[0];            // += partials later
}

// ---- one LSTM time step ---------------------------------------------------
// grid = 128 blocks (64 j-tiles x 2 batch-halves), 128 threads = 4 waves.
// Each wave: one 16(j) x 16(b) tile across ALL 4 gate quadrants, K=1024.
__global__ __launch_bounds__(128) void lstm_step_kernel(
    const float* __restrict__ Apack, const float* __restrict__ combo,
    const float* __restrict__ hT_in, float* __restrict__ hT_out,
    float* __restrict__ cT, const float* __restrict__ W_lin,
    float* __restrict__ out, int t)
{
    const int wave  = threadIdx.x >> 5;
    const int lane  = threadIdx.x & 31;
    const int lo    = lane & 15;
    const int hi    = lane >> 4;
    const int jtile = blockIdx.x >> 1;                 // 0..63
    const int btile = (blockIdx.x & 1) * 4 + wave;     // 0..7
    const int j0    = jtile * 16;
    const int b0    = btile * 16;

    v8f acc0 = {0.f,0.f,0.f,0.f,0.f,0.f,0.f,0.f};
    v8f acc1 = acc0, acc2 = acc0, acc3 = acc0;

    // A: coalesced b128 per quadrant per k-pair; B: coalesced b64 per k-step.
    const float* ap = Apack + (size_t)jtile * (128 * 512) + lane * 4;
    const float* bp = hT_in + b0 * 4 + lo * 4 + hi * 2;

#pragma unroll 4
    for (int kp = 0; kp < 128; ++kp) {
        v4f a0 = *(const v4f*)(ap +   0);   // quadrant i
        v4f a1 = *(const v4f*)(ap + 128);   // quadrant f
        v4f a2 = *(const v4f*)(ap + 256);   // quadrant g
        v4f a3 = *(const v4f*)(ap + 384);   // quadrant o
        v2f bx = *(const v2f*)(bp);         // k-step 2*kp
        v2f by = *(const v2f*)(bp + 512);   // k-step 2*kp+1
        ap += 512;
        bp += 1024;
        acc0 = wmma4((v2f){a0.x, a0.y}, bx, acc0);
        acc1 = wmma4((v2f){a1.x, a1.y}, bx, acc1);
        acc2 = wmma4((v2f){a2.x, a2.y}, bx, acc2);
        acc3 = wmma4((v2f){a3.x, a3.y}, bx, acc3);
        acc0 = wmma4((v2f){a0.z, a0.w}, by, acc0);
        acc1 = wmma4((v2f){a1.z, a1.w}, by, acc1);
        acc2 = wmma4((v2f){a2.z, a2.w}, by, acc2);
        acc3 = wmma4((v2f){a3.z, a3.w}, by, acc3);
    }

    // ---- fused LSTM pointwise + linear head (all in registers) ----
    const int b = b0 + lo;
    float prev = 0.0f, flag = 0.0f;
    if (t > 0) { prev = out[b * TSTEPS + (t - 1)]; flag = 1.0f; }  // feedback

    float s = 0.0f;
#pragma unroll
    for (int v = 0; v < 8; ++v) {
        const int j = j0 + v + 8 * hi;                 // D layout M index
        v4f ci = *(const v4f*)(combo + 4 * (0 * HID + j));
        v4f cf = *(const v4f*)(combo + 4 * (1 * HID + j));
        v4f cg = *(const v4f*)(combo + 4 * (2 * HID + j));
        v4f co = *(const v4f*)(combo + 4 * (3 * HID + j));
        float gi = acc0[v] + ci.x + prev * ci.y + flag * ci.z;
        float gf = acc1[v] + cf.x + prev * cf.y + flag * cf.z;
        float gg = acc2[v] + cg.x + prev * cg.y + flag * cg.z;
        float go = acc3[v] + co.x + prev * co.y + flag * co.z;

        float cold = cT[j * BATCH + b];
        float cn   = sigm(gf) * cold + sigm(gi) * tanhf(gg);
        cT[j * BATCH + b] = cn;
        float hv   = sigm(go) * tanhf(cn);

        hT_out[(j >> 2) * (BATCH * 4) + b * 4 + (j & 3)] = hv;  // next-step B
        s += hv * W_lin[j];                                      // head partial
    }
    atomicAdd(&out[b * TSTEPS + t], s);   // d_out pre-seeded with b_lin
}

// ---------------------------------------------------------------------------
extern "C" void kernel_launch(void* const* d_in, const int* in_sizes, int n_in,
                              void* d_out, int out_size, void* d_ws, size_t ws_size,
                              hipStream_t stream) {
    const float* h0    = (const float*)d_in[0];
    const float* c0    = (const float*)d_in[1];
    const float* W_ih  = (const float*)d_in[2];
    const float* W_hh  = (const float*)d_in[3];
    const float* b_ih  = (const float*)d_in[4];
    const float* b_hh  = (const float*)d_in[5];
    const float* W_lin = (const float*)d_in[6];
    const float* b_lin = (const float*)d_in[7];
    float* out = (float*)d_out;

    float* ws    = (float*)d_ws;          // ~18.4 MB of f32 scratch
    float* Apack = ws;                    // 4096*1024
    float* hTa   = Apack + 4194304;       // 1024*128
    float* hTb   = hTa + 131072;          // 1024*128
    float* cT    = hTb + 131072;          // 1024*128
    float* combo = cT + 131072;           // 4096*4

    repack_kernel<<<4096, 256, 0, stream>>>(W_hh, W_ih, b_ih, b_hh, Apack, combo);
    init_state_kernel<<<512, 256, 0, stream>>>(h0, c0, b_lin, hTa, cT, out);

    for (int t = 0; t < TSTEPS; ++t) {
        float* hin  = (t & 1) ? hTb : hTa;
        float* hout = (t & 1) ? hTa : hTb;
        lstm_step_kernel<<<128, 128, 0, stream>>>(Apack, combo, hin, hout,
                                                  cT, W_lin, out, t);
    }
}